// GCN_46402826666261
// MI455X (gfx1250) — compile-verified
//
#include <hip/hip_runtime.h>

typedef float v2f __attribute__((ext_vector_type(2)));
typedef float v8f __attribute__((ext_vector_type(8)));

#define N_NODES 100000
#define IN_DIM  8
#define HIDDEN  128
#define OUT_DIM 16
#define N_EDGES 3200000

// ---------------- degree / normalization ----------------

__global__ void gcn_init_deg(float* __restrict__ deg) {
    int i = blockIdx.x * blockDim.x + threadIdx.x;
    if (i < N_NODES) deg[i] = 1.0f;  // self-loop contributes 1
}

__global__ void gcn_accum_deg(const int* __restrict__ dst, float* __restrict__ deg) {
    int i = blockIdx.x * blockDim.x + threadIdx.x;
    if (i < N_EDGES) atomicAdd(&deg[dst[i]], 1.0f);
}

__global__ void gcn_dinv(float* __restrict__ deg) {
    int i = blockIdx.x * blockDim.x + threadIdx.x;
    if (i < N_NODES) deg[i] = rsqrtf(deg[i]);  // deg >= 1 always (self loops)
}

// ---------------- dense GEMMs via f32 WMMA ----------------
// Layer 1: xw = x[N,8] @ W1[8,128].  One wave per 16x16 output tile.
// Block = 256 threads = 8 waves -> the 8 column tiles of one 16-row stripe.
__global__ __launch_bounds__(256) void gcn_gemm1(const float* __restrict__ x,
                                                 const float* __restrict__ W1,
                                                 float* __restrict__ xw) {
    const int wave = threadIdx.x >> 5;
    const int lane = threadIdx.x & 31;
    const int lr   = lane & 15;   // row-within-tile index (A) / column index (B,C)
    const int hi   = lane >> 4;   // half-wave select
    const int row0 = blockIdx.x * 16;
    const int col0 = wave * 16;

    v8f c = {};
#pragma unroll
    for (int ks = 0; ks < 2; ++ks) {          // K = 8 -> two 16x16x4 steps
        const int kb = ks * 4 + 2 * hi;       // A: lanes0-15 hold K=kb,kb+1; lanes16-31 K=kb+2.. per ISA layout
        v2f a, b;
        a.x = x[(row0 + lr) * IN_DIM + kb];
        a.y = x[(row0 + lr) * IN_DIM + kb + 1];
        b.x = W1[kb * HIDDEN + col0 + lr];
        b.y = W1[(kb + 1) * HIDDEN + col0 + lr];
        c = __builtin_amdgcn_wmma_f32_16x16x4_f32(false, a, false, b, (short)0, c, false, false);
    }
#pragma unroll
    for (int r = 0; r < 8; ++r)               // C/D: vgpr r, half-wave hi -> M = r + 8*hi, N = lr
        xw[(size_t)(row0 + r + 8 * hi) * HIDDEN + col0 + lr] = c[r];
}

// Layer 2: hw = h[N,128] @ W2[128,16].  One wave per 16x16 tile, 32 k-steps.
__global__ __launch_bounds__(256) void gcn_gemm2(const float* __restrict__ h,
                                                 const float* __restrict__ W2,
                                                 float* __restrict__ hw) {
    const int wave = threadIdx.x >> 5;
    const int lane = threadIdx.x & 31;
    const int lr   = lane & 15;
    const int hi   = lane >> 4;
    const int tile = blockIdx.x * 8 + wave;   // 6250 tiles total
    if (tile >= N_NODES / 16) return;         // wave-uniform: EXEC stays all-ones for active waves
    const int row0 = tile * 16;

    v8f c = {};
    for (int kb0 = 0; kb0 < HIDDEN; kb0 += 4) {
        const int kb = kb0 + 2 * hi;
        v2f a, b;
        a.x = h[(size_t)(row0 + lr) * HIDDEN + kb];
        a.y = h[(size_t)(row0 + lr) * HIDDEN + kb + 1];
        b.x = W2[kb * OUT_DIM + lr];
        b.y = W2[(kb + 1) * OUT_DIM + lr];
        c = __builtin_amdgcn_wmma_f32_16x16x4_f32(false, a, false, b, (short)0, c, false, false);
    }
#pragma unroll
    for (int r = 0; r < 8; ++r)
        hw[(size_t)(row0 + r + 8 * hi) * OUT_DIM + lr] = c[r];
}

// ---------------- sparse scatter (L2-resident f32 atomics) ----------------
// Layer 1: 32 lanes per edge, float4 per lane (128 features).
__global__ void gcn_scatter1(const int* __restrict__ src, const int* __restrict__ dst,
                             const float* __restrict__ dinv, const float* __restrict__ xw,
                             float* __restrict__ agg) {
    long long t = (long long)blockIdx.x * blockDim.x + threadIdx.x;
    const long long total = (long long)(N_EDGES + N_NODES) * 32;
    if (t >= total) return;
    const int e    = (int)(t >> 5);
    const int lane = (int)(t & 31);
    int s, d;
    if (e < N_EDGES) { s = src[e]; d = dst[e]; }
    else             { s = d = e - N_EDGES; }   // self loop
    const float w = dinv[s] * dinv[d];
    const float4 v = ((const float4*)(xw + (size_t)s * HIDDEN))[lane];
    float* o = agg + (size_t)d * HIDDEN + lane * 4;
    atomicAdd(o + 0, v.x * w);
    atomicAdd(o + 1, v.y * w);
    atomicAdd(o + 2, v.z * w);
    atomicAdd(o + 3, v.w * w);
}

// Layer 2: 4 lanes per edge (16 features).
__global__ void gcn_scatter2(const int* __restrict__ src, const int* __restrict__ dst,
                             const float* __restrict__ dinv, const float* __restrict__ hw,
                             float* __restrict__ agg) {
    long long t = (long long)blockIdx.x * blockDim.x + threadIdx.x;
    const long long total = (long long)(N_EDGES + N_NODES) * 4;
    if (t >= total) return;
    const int e    = (int)(t >> 2);
    const int lane = (int)(t & 3);
    int s, d;
    if (e < N_EDGES) { s = src[e]; d = dst[e]; }
    else             { s = d = e - N_EDGES; }
    const float w = dinv[s] * dinv[d];
    const float4 v = ((const float4*)(hw + (size_t)s * OUT_DIM))[lane];
    float* o = agg + (size_t)d * OUT_DIM + lane * 4;
    atomicAdd(o + 0, v.x * w);
    atomicAdd(o + 1, v.y * w);
    atomicAdd(o + 2, v.z * w);
    atomicAdd(o + 3, v.w * w);
}

// ---------------- epilogues ----------------

__global__ void gcn_bias_relu(float* __restrict__ h, const float* __restrict__ b1) {
    int i = blockIdx.x * blockDim.x + threadIdx.x;
    if (i < N_NODES * HIDDEN) {
        float v = h[i] + b1[i & (HIDDEN - 1)];
        h[i] = v > 0.0f ? v : 0.0f;
    }
}

__global__ void gcn_bias_out(const float* __restrict__ agg, const float* __restrict__ b2,
                             float* __restrict__ out) {
    int i = blockIdx.x * blockDim.x + threadIdx.x;
    if (i < N_NODES * OUT_DIM) out[i] = agg[i] + b2[i & (OUT_DIM - 1)];
}

// ---------------- launcher ----------------

extern "C" void kernel_launch(void* const* d_in, const int* in_sizes, int n_in,
                              void* d_out, int out_size, void* d_ws, size_t ws_size,
                              hipStream_t stream) {
    const float* x  = (const float*)d_in[0];
    const int*   ei = (const int*)d_in[1];     // [2, E] row-major, int32
    const float* W1 = (const float*)d_in[2];   // [8, 128]
    const float* b1 = (const float*)d_in[3];   // [128]
    const float* W2 = (const float*)d_in[4];   // [128, 16]
    const float* b2 = (const float*)d_in[5];   // [16]
    const int* src = ei;
    const int* dst = ei + N_EDGES;

    float* ws   = (float*)d_ws;
    float* dinv = ws;                                    // N floats
    float* xw   = ws + N_NODES;                          // N*128 floats
    float* agg1 = xw + (size_t)N_NODES * HIDDEN;         // N*128 floats (becomes h in place)
    // xw region is dead after scatter1; reuse it for layer-2 buffers:
    float* hw   = xw;                                    // N*16 floats
    float* agg2 = xw + (size_t)N_NODES * OUT_DIM;        // N*16 floats

    // 1) symmetric normalization weights
    gcn_init_deg <<<(N_NODES + 255) / 256, 256, 0, stream>>>(dinv);
    gcn_accum_deg<<<(N_EDGES + 255) / 256, 256, 0, stream>>>(dst, dinv);
    gcn_dinv     <<<(N_NODES + 255) / 256, 256, 0, stream>>>(dinv);

    // 2) layer 1: GEMM (WMMA f32) -> scatter -> bias+relu
    hipMemsetAsync(agg1, 0, (size_t)N_NODES * HIDDEN * sizeof(float), stream);
    gcn_gemm1<<<N_NODES / 16, 256, 0, stream>>>(x, W1, xw);
    {
        long long total = (long long)(N_EDGES + N_NODES) * 32;
        gcn_scatter1<<<(unsigned)((total + 255) / 256), 256, 0, stream>>>(src, dst, dinv, xw, agg1);
    }
    gcn_bias_relu<<<(N_NODES * HIDDEN + 255) / 256, 256, 0, stream>>>(agg1, b1);

    // 3) layer 2: GEMM (WMMA f32) -> scatter -> bias
    hipMemsetAsync(agg2, 0, (size_t)N_NODES * OUT_DIM * sizeof(float), stream);
    gcn_gemm2<<<(N_NODES / 16 + 7) / 8, 256, 0, stream>>>(agg1, W2, hw);
    {
        long long total = (long long)(N_EDGES + N_NODES) * 4;
        gcn_scatter2<<<(unsigned)((total + 255) / 256), 256, 0, stream>>>(src, dst, dinv, hw, agg2);
    }
    gcn_bias_out<<<(N_NODES * OUT_DIM + 255) / 256, 256, 0, stream>>>(agg2, b2, (float*)d_out);
}